// NovelViewProjector_9801115370009
// MI455X (gfx1250) — compile-verified
//
#include <hip/hip_runtime.h>

typedef __attribute__((ext_vector_type(2))) float v2f;
typedef __attribute__((ext_vector_type(8))) float v8f;

#define NVERT   289
#define NFACE   512
#define IMG     128
#define NPIX    (IMG*IMG)
#define KF      8
#define SIGMA_  1e-4f
#define GAMMA_  1e-4f
#define BLUR_   1e-4f
#define ZFAR_   100.0f
#define ZNEAR_  1.0f
#define ZEPS_   0.01f
#define FSTRIDE 16   // floats per face record

// ---- gfx1250 f32 WMMA availability (device pass only; host pass takes scalar path)
#if defined(__HIP_DEVICE_COMPILE__) && __has_builtin(__builtin_amdgcn_wmma_f32_16x16x4_f32)
#define HAVE_WMMA_F32X4 1
#else
#define HAVE_WMMA_F32X4 0
#endif

// ------------------------------------------------------------------
// Kernel 1: vertex transform -> screen space  (ws: NVERT x 4 floats)
// ------------------------------------------------------------------
__global__ void nvp_verts(const float* __restrict__ xy_off,
                          const float* __restrict__ z_grid,
                          const float* __restrict__ R_in,
                          const float* __restrict__ T_in,
                          const float* __restrict__ R_out,
                          const float* __restrict__ T_out,
                          float* __restrict__ verts) {
  int v = blockIdx.x * blockDim.x + threadIdx.x;
  if (v >= NVERT) return;
  int ny = v / 17, nx = v % 17;
  const float S = 0.57735026918962576451f;        // tan(30 deg)
  float gx = -1.0f + 0.125f * (float)nx;          // linspace(-1,1,17)
  float gy = -1.0f + 0.125f * (float)ny;
  float sx = (gx + xy_off[2*v+0]) * S;
  float sy = (gy + xy_off[2*v+1]) * S;
  float z  = z_grid[v];
  float p0 = sx * z, p1 = sy * z, p2 = z;         // scaled_verts
  float q0 = p0 - T_in[0], q1 = p1 - T_in[1], q2 = p2 - T_in[2];
  // world_k = sum_j q_j * R_in[k][j]
  float w0 = q0*R_in[0] + q1*R_in[1] + q2*R_in[2];
  float w1 = q0*R_in[3] + q1*R_in[4] + q2*R_in[5];
  float w2 = q0*R_in[6] + q1*R_in[7] + q2*R_in[8];
  // view_k = sum_j w_j * R_out[j][k] + T_out[k]
  float vx = w0*R_out[0] + w1*R_out[3] + w2*R_out[6] + T_out[0];
  float vy = w0*R_out[1] + w1*R_out[4] + w2*R_out[7] + T_out[1];
  float vz = w0*R_out[2] + w1*R_out[5] + w2*R_out[8] + T_out[2];
  float zden = (vz >= 0.0f) ? fmaxf(vz, ZEPS_) : fminf(vz, -ZEPS_);
  float s = 1.0f / S;
  verts[4*v+0] = s * vx / zden;
  verts[4*v+1] = s * vy / zden;
  verts[4*v+2] = vz;
  verts[4*v+3] = 0.0f;
}

// ------------------------------------------------------------------
// Kernel 2: per-face record (ws: NFACE x 16 floats)
// [ax,ay,bx,by,cx,cy, z0,z1,z2, inv_area_s, u0,v0,u1,v1,u2,v2]
// ------------------------------------------------------------------
__global__ void nvp_faces(const float* __restrict__ verts,
                          float* __restrict__ faces) {
  int f = blockIdx.x * blockDim.x + threadIdx.x;
  if (f >= NFACE) return;
  int g = f >> 6, i = f & 63;
  int ny = 2*(i >> 3) + ((g >= 4) ? 1 : 0);
  int nx = 2*(i & 7)  + (((g < 2) || (g >= 6)) ? 1 : 0);
  int pat = g & 3;
  int y0, x0, y1, x1, y2, x2;
  if      (pat == 0) { y0=ny;   x0=nx+1; y1=ny;   x1=nx;   y2=ny+1; x2=nx;   }
  else if (pat == 1) { y0=ny;   x0=nx+1; y1=ny+1; x1=nx;   y2=ny+1; x2=nx+1; }
  else if (pat == 2) { y0=ny;   x0=nx+1; y1=ny;   x1=nx;   y2=ny+1; x2=nx+1; }
  else               { y0=ny+1; x0=nx+1; y1=ny;   x1=nx;   y2=ny+1; x2=nx;   }
  int i0 = y0*17 + x0, i1 = y1*17 + x1, i2 = y2*17 + x2;
  float ax = verts[4*i0+0], ay = verts[4*i0+1], z0 = verts[4*i0+2];
  float bx = verts[4*i1+0], by = verts[4*i1+1], z1 = verts[4*i1+2];
  float cx = verts[4*i2+0], cy = verts[4*i2+1], z2 = verts[4*i2+2];
  float area = (bx-ax)*(cy-ay) - (by-ay)*(cx-ax);
  float area_s = (fabsf(area) < 1e-8f) ? 1e-8f : area;
  float* F = &faces[f * FSTRIDE];
  F[0]=ax; F[1]=ay; F[2]=bx; F[3]=by; F[4]=cx; F[5]=cy;
  F[6]=z0; F[7]=z1; F[8]=z2; F[9]=1.0f/area_s;
  F[10] = 1.0f - 0.0625f*(float)x0;  F[11] = 0.0625f*(float)y0;
  F[12] = 1.0f - 0.0625f*(float)x1;  F[13] = 0.0625f*(float)y1;
  F[14] = 1.0f - 0.0625f*(float)x2;  F[15] = 0.0625f*(float)y2;
}

// point-to-segment squared distance (matches reference seg_d2)
__device__ __forceinline__ float segd2(float px, float py,
                                       float sx, float sy,
                                       float ex, float ey) {
  float dx = ex - sx, dy = ey - sy;
  float dd = fmaxf(dx*dx + dy*dy, 1e-12f);
  float t  = ((px - sx)*dx + (py - sy)*dy) / dd;
  t = fminf(fmaxf(t, 0.0f), 1.0f);
  float qx = sx + t*dx - px, qy = sy + t*dy - py;
  return qx*qx + qy*qy;
}

// ------------------------------------------------------------------
// Kernel 3: soft rasterize + shade. One wave (32 lanes) = 32 pixels
// of one image row. Edge functions for 16px x 16face tiles computed
// on the matrix unit via V_WMMA_F32_16X16X4_F32.
// ------------------------------------------------------------------
__launch_bounds__(32)
__global__ void nvp_raster(const float* __restrict__ facesg,
                           const float* __restrict__ tex,
                           float* __restrict__ out) {
  __shared__ float sF[NFACE * FSTRIDE];   // 32 KB face table
  __shared__ float sW[3][32][17];         // edge values, padded stride 17

  const int lane = threadIdx.x;
  const int base = blockIdx.x << 5;       // first pixel of this wave

  // cache face table in LDS (128-bit copies)
  for (int t = lane; t < (NFACE*FSTRIDE)/4; t += 32)
    ((float4*)sF)[t] = ((const float4*)facesg)[t];
  __syncthreads();

  const int   pid = base + lane;
  const float px  = 1.0f - 2.0f*((float)(pid & (IMG-1)) + 0.5f)/(float)IMG;
  const float py  = 1.0f - 2.0f*((float)(pid >> 7)      + 0.5f)/(float)IMG; // row-constant

  const int  m  = lane & 15;
  const bool hi = lane >= 16;
  const float pxA = 1.0f - 2.0f*((float)((base + m)      & (IMG-1)) + 0.5f)/(float)IMG;
  const float pxB = 1.0f - 2.0f*((float)((base + 16 + m) & (IMG-1)) + 0.5f)/(float)IMG;

  // A matrices (16x4 f32): lanes 0-15 carry K={0,1}=(1,px), lanes 16-31 K={2,3}=(py,0)
  v2f A0, A1;
  A0.x = hi ? py : 1.0f;  A0.y = hi ? 0.0f : pxA;
  A1.x = hi ? py : 1.0f;  A1.y = hi ? 0.0f : pxB;

  float key8[KF]; int idx8[KF];
  #pragma unroll
  for (int k = 0; k < KF; ++k) { key8[k] = __builtin_inff(); idx8[k] = -1; }

  for (int chunk = 0; chunk < NFACE/16; ++chunk) {
    const int fb = chunk << 4;
    // face handled by this lane's B column
    const float* Fb = &sF[(fb + m) * FSTRIDE];
    const float ax = Fb[0], ay = Fb[1], bx = Fb[2], by = Fb[3], cx = Fb[4], cy = Fb[5];
    // edge(p,q,r) = (qx*ry - qy*rx) + px*(qy-ry) + py*(rx-qx)
    const float cc0 = bx*cy - by*cx, kx0 = by - cy, ky0 = cx - bx; // edge(p,b,c)
    const float cc1 = cx*ay - cy*ax, kx1 = cy - ay, ky1 = ax - cx; // edge(p,c,a)
    const float cc2 = ax*by - ay*bx, kx2 = ay - by, ky2 = bx - ax; // edge(p,a,b)
    const int pr = hi ? 8 : 0;

#if HAVE_WMMA_F32X4
    // B matrices (4x16 f32): lanes 0-15 K={0,1}=(cc,kx), lanes 16-31 K={2,3}=(ky,0)
    v2f B0, B1, B2;
    B0.x = hi ? ky0 : cc0;  B0.y = hi ? 0.0f : kx0;
    B1.x = hi ? ky1 : cc1;  B1.y = hi ? 0.0f : kx1;
    B2.x = hi ? ky2 : cc2;  B2.y = hi ? 0.0f : kx2;
    const v8f Z = {0.f,0.f,0.f,0.f,0.f,0.f,0.f,0.f};
    v8f D00 = __builtin_amdgcn_wmma_f32_16x16x4_f32(false, A0, false, B0, (short)0, Z, false, false);
    v8f D01 = __builtin_amdgcn_wmma_f32_16x16x4_f32(false, A0, false, B1, (short)0, Z, false, false);
    v8f D02 = __builtin_amdgcn_wmma_f32_16x16x4_f32(false, A0, false, B2, (short)0, Z, false, false);
    v8f D10 = __builtin_amdgcn_wmma_f32_16x16x4_f32(false, A1, false, B0, (short)0, Z, false, false);
    v8f D11 = __builtin_amdgcn_wmma_f32_16x16x4_f32(false, A1, false, B1, (short)0, Z, false, false);
    v8f D12 = __builtin_amdgcn_wmma_f32_16x16x4_f32(false, A1, false, B2, (short)0, Z, false, false);
    // D tile: lane holds rows {pr..pr+7} of column m; scatter to sW[edge][pixel][face]
    #pragma unroll
    for (int r = 0; r < 8; ++r) {
      sW[0][pr + r][m]      = D00[r];
      sW[1][pr + r][m]      = D01[r];
      sW[2][pr + r][m]      = D02[r];
      sW[0][16 + pr + r][m] = D10[r];
      sW[1][16 + pr + r][m] = D11[r];
      sW[2][16 + pr + r][m] = D12[r];
    }
#else
    // scalar fallback: identical data layout
    #pragma unroll
    for (int r = 0; r < 8; ++r) {
      #pragma unroll
      for (int tl = 0; tl < 2; ++tl) {
        int pixr = tl*16 + pr + r;
        float pxp = 1.0f - 2.0f*((float)((base + pixr) & (IMG-1)) + 0.5f)/(float)IMG;
        sW[0][pixr][m] = cc0 + kx0*pxp + ky0*py;
        sW[1][pixr][m] = cc1 + kx1*pxp + ky1*py;
        sW[2][pixr][m] = cc2 + kx2*pxp + ky2*py;
      }
    }
#endif
    __syncthreads();

    // each lane consumes its own pixel's 16 candidates
    #pragma unroll 4
    for (int n = 0; n < 16; ++n) {
      const int f = fb + n;
      const float* G = &sF[f * FSTRIDE];
      float w0 = sW[0][lane][n], w1 = sW[1][lane][n], w2 = sW[2][lane][n];
      float inva = G[9];
      float b0 = w0*inva, b1 = w1*inva, b2 = w2*inva;
      bool inside = (b0 >= 0.0f) && (b1 >= 0.0f) && (b2 >= 0.0f);
      float z0 = G[6], z1 = G[7], z2 = G[8];
      float n0 = b0*z1*z2, n1 = b1*z0*z2, n2 = b2*z0*z1;
      float den = n0 + n1 + n2;
      den = (fabsf(den) < 1e-10f) ? 1e-10f : den;
      float q0 = fmaxf(n0/den, 0.0f);
      float q1 = fmaxf(n1/den, 0.0f);
      float q2 = fmaxf(n2/den, 0.0f);
      float cs = fmaxf(q0 + q1 + q2, 1e-10f);
      float zpix = (q0*z0 + q1*z1 + q2*z2) / cs;
      float gax=G[0], gay=G[1], gbx=G[2], gby=G[3], gcx=G[4], gcy=G[5];
      float d2 = fminf(fminf(segd2(px,py,gax,gay,gbx,gby),
                             segd2(px,py,gbx,gby,gcx,gcy)),
                             segd2(px,py,gcx,gcy,gax,gay));
      float sd = inside ? -d2 : d2;
      bool valid = (sd < BLUR_) && (zpix > 1e-4f);
      float key = valid ? zpix : __builtin_inff();
      if (key < key8[KF-1]) {                 // stable top-8 by (z, face index)
        key8[KF-1] = key; idx8[KF-1] = f;
        #pragma unroll
        for (int j = KF-1; j > 0; --j) {
          if (key8[j] < key8[j-1]) {
            float tk = key8[j]; key8[j] = key8[j-1]; key8[j-1] = tk;
            int   ti = idx8[j]; idx8[j] = idx8[j-1]; idx8[j-1] = ti;
          }
        }
      }
    }
    __syncthreads();
  }

  // -------- shading of the K selected faces --------
  float prob[KF], zinv[KF], zks[KF], cr[KF], cg[KF], cb[KF];
  #pragma unroll
  for (int k = 0; k < KF; ++k) {
    int f = idx8[k];
    if (f < 0) { prob[k]=0.f; zinv[k]=0.f; zks[k]=ZFAR_; cr[k]=cg[k]=cb[k]=0.f; continue; }
    const float* G = &sF[f * FSTRIDE];
    float ax=G[0], ay=G[1], bx=G[2], by=G[3], cx=G[4], cy=G[5];
    float z0=G[6], z1=G[7], z2=G[8];
    float w0 = (bx*cy - by*cx) + (by-cy)*px + (cx-bx)*py;
    float w1 = (cx*ay - cy*ax) + (cy-ay)*px + (ax-cx)*py;
    float w2 = (ax*by - ay*bx) + (ay-by)*px + (bx-ax)*py;
    float inva = G[9];
    float b0 = w0*inva, b1 = w1*inva, b2 = w2*inva;
    bool inside = (b0 >= 0.0f) && (b1 >= 0.0f) && (b2 >= 0.0f);
    float n0 = b0*z1*z2, n1 = b1*z0*z2, n2 = b2*z0*z1;
    float den = n0 + n1 + n2;
    den = (fabsf(den) < 1e-10f) ? 1e-10f : den;
    float q0 = fmaxf(n0/den, 0.0f);
    float q1 = fmaxf(n1/den, 0.0f);
    float q2 = fmaxf(n2/den, 0.0f);
    float cs = fmaxf(q0 + q1 + q2, 1e-10f);
    float zpix = (q0*z0 + q1*z1 + q2*z2) / cs;
    float d2 = fminf(fminf(segd2(px,py,ax,ay,bx,by),
                           segd2(px,py,bx,by,cx,cy)),
                           segd2(px,py,cx,cy,ax,ay));
    float sd = inside ? -d2 : d2;
    bool valid = (sd < BLUR_) && (zpix > 1e-4f);
    float vk = valid ? 1.0f : 0.0f;
    // interpolated uv, bilinear texture fetch
    float iu = (q0*G[10] + q1*G[12] + q2*G[14]) / cs;
    float iv = (q0*G[11] + q1*G[13] + q2*G[15]) / cs;
    float tx = iu * 127.0f;
    float ty = (1.0f - iv) * 127.0f;
    float x0f = fminf(fmaxf(floorf(tx), 0.0f), 127.0f);
    float y0f = fminf(fmaxf(floorf(ty), 0.0f), 127.0f);
    int ix0 = (int)x0f, iy0 = (int)y0f;
    int ix1 = (ix0 + 1 < 127) ? ix0 + 1 : 127;
    int iy1 = (iy0 + 1 < 127) ? iy0 + 1 : 127;
    float wx = tx - x0f, wy = ty - y0f;
    const float* t00 = &tex[(iy0*IMG + ix0)*3];
    const float* t01 = &tex[(iy0*IMG + ix1)*3];
    const float* t10 = &tex[(iy1*IMG + ix0)*3];
    const float* t11 = &tex[(iy1*IMG + ix1)*3];
    float col[3];
    #pragma unroll
    for (int c = 0; c < 3; ++c) {
      float top = (1.0f-wx)*t00[c] + wx*t01[c];
      float bot = (1.0f-wx)*t10[c] + wx*t11[c];
      col[c] = (1.0f-wy)*top + wy*bot;
    }
    cr[k] = col[0]; cg[k] = col[1]; cb[k] = col[2];
    prob[k] = vk / (1.0f + expf(sd / SIGMA_));       // sigmoid(-sd/sigma)*vk
    float zz = valid ? zpix : ZFAR_;
    zks[k]  = zz;
    zinv[k] = (ZFAR_ - zz) / (ZFAR_ - ZNEAR_) * vk;
  }

  float zmax = -__builtin_inff();
  #pragma unroll
  for (int k = 0; k < KF; ++k) zmax = fmaxf(zmax, zinv[k]);
  zmax = fmaxf(zmax, 1e-10f);

  float swn = 0.f, sr = 0.f, sg = 0.f, sb = 0.f, sz = 0.f, keep = 1.f;
  #pragma unroll
  for (int k = 0; k < KF; ++k) {
    float wn = prob[k] * expf((zinv[k] - zmax) / GAMMA_);
    swn += wn; sr += wn*cr[k]; sg += wn*cg[k]; sb += wn*cb[k]; sz += wn*zks[k];
    keep *= (1.0f - prob[k]);
  }
  float delta = expf((1e-10f - zmax) / GAMMA_);
  float denom = swn + delta;
  float* o = &out[pid * 5];
  o[0] = sr / denom;
  o[1] = sg / denom;
  o[2] = sb / denom;                 // BG color is 0
  o[3] = 1.0f - keep;
  o[4] = (sz + delta * ZFAR_) / denom;
}

// ------------------------------------------------------------------
extern "C" void kernel_launch(void* const* d_in, const int* in_sizes, int n_in,
                              void* d_out, int out_size, void* d_ws, size_t ws_size,
                              hipStream_t stream) {
  (void)in_sizes; (void)n_in; (void)out_size; (void)ws_size;
  const float* xy   = (const float*)d_in[0];  // (1,289,2)
  const float* zg   = (const float*)d_in[1];  // (1,289,1)
  const float* tex  = (const float*)d_in[2];  // (1,128,128,3)
  const float* Rin  = (const float*)d_in[3];  // (1,3,3)
  const float* Tin  = (const float*)d_in[4];  // (1,3)
  const float* Rout = (const float*)d_in[5];  // (1,3,3)
  const float* Tout = (const float*)d_in[6];  // (1,3)

  float* ws    = (float*)d_ws;
  float* verts = ws;               // NVERT*4 floats
  float* faces = ws + 2048;        // byte offset 8192, NFACE*16 floats

  nvp_verts<<<(NVERT + 63)/64, 64, 0, stream>>>(xy, zg, Rin, Tin, Rout, Tout, verts);
  nvp_faces<<<(NFACE + 63)/64, 64, 0, stream>>>(verts, faces);
  nvp_raster<<<NPIX/32, 32, 0, stream>>>(faces, tex, (float*)d_out);
}